// HashGridEncodeTorch_40596030882166
// MI455X (gfx1250) — compile-verified
//
#include <hip/hip_runtime.h>

// ---------------------------------------------------------------------------
// HashGridEncode (instant-NGP style) for MI455X / gfx1250.
//
// MI455X-specific choices:
//  * All 16 hash tables total 64 MB -> resident in the 192 MB L2. Gathers use
//    default (RT) temporal policy so they stay cached.
//  * The 256 MB output stream uses non-temporal stores (TH=NT) so it does NOT
//    evict the table working set from L2.
//  * One global_load_b64 per corner (FEAT=2 floats packed), 4 corners x 16
//    levels fully unrolled for ILP.
//  * Bilinear lerp done on 2-wide float vectors so the backend emits packed
//    FP32 VOP3P ops (v_pk_mul_f32 / v_pk_fma_f32): 6 packed ops per level
//    instead of 12 scalar ones.
//  * All 32 outputs kept in VGPRs, written as 8 x non-temporal
//    global_store_b128 (each lane writes 128 contiguous bytes = whole lines).
// ---------------------------------------------------------------------------

typedef float v2f __attribute__((ext_vector_type(2)));
typedef float v4f __attribute__((ext_vector_type(4)));

static constexpr int      kLevels    = 16;
static constexpr unsigned kTableSize = 1u << 19;          // 2^LOG2_T
static constexpr unsigned kMask      = kTableSize - 1u;
static constexpr unsigned kPI1       = 2654435761u;
static constexpr int      kFeat      = 2;
static constexpr int      kOutF      = kLevels * kFeat;   // 32 floats / point

// Per-level resolution: int(B_SCALE^i * START_RES), computed at compile time.
struct ResTable { float r[kLevels]; };
static constexpr ResTable make_res() {
  ResTable t{};
  double b = 1.0;
  const double kB = 1.447269237440378;
  for (int i = 0; i < kLevels; ++i) {
    t.r[i] = (float)(long)(b * 16.0);
    b *= kB;
  }
  return t;
}
static constexpr ResTable kRes = make_res();

__global__ __launch_bounds__(256)
void hashgrid_encode_kernel(const float2* __restrict__ x,
                            const v2f*    __restrict__ tables,
                            v4f*          __restrict__ out,
                            int n)
{
  const int p = blockIdx.x * blockDim.x + threadIdx.x;
  if (p >= n) return;

  const float2 xy = x[p];          // coalesced b64 load, scale_offset

  v2f o[kLevels];                  // 32 output floats, register-resident

#pragma unroll
  for (int i = 0; i < kLevels; ++i) {
    const float res = kRes.r[i];   // compile-time constant per iteration

    const float xs = xy.x * res;
    const float ys = xy.y * res;
    const float rx = floorf(xs);
    const float ry = floorf(ys);
    const float dx = xs - rx;
    const float dy = ys - ry;

    const unsigned ix = (unsigned)(int)rx;      // x in [0,1): non-negative
    const unsigned iy = (unsigned)(int)ry;

    // h(vx,vy) = (vx * 1) ^ (vy * PI1); table is 2^19 entries -> AND mask.
    const unsigned px0 = ix;
    const unsigned px1 = ix + 1u;
    const unsigned py0 = iy * kPI1;
    const unsigned py1 = py0 + kPI1;            // (iy+1)*PI1 reuses the mul

    const v2f* __restrict__ t = tables + (size_t)i * kTableSize;

    // 4 corner gathers, one b64 each (both features). L2-resident tables.
    const v2f e00 = t[(px0 ^ py0) & kMask];     // (x0, y0)
    const v2f e01 = t[(px0 ^ py1) & kMask];     // (x0, y1)
    const v2f e10 = t[(px1 ^ py0) & kMask];     // (x1, y0)
    const v2f e11 = t[(px1 ^ py1) & kMask];     // (x1, y1)

    const float omx = 1.0f - dx;
    const float omy = 1.0f - dy;

    // Packed FP32 lerps over the FEAT=2 axis:
    //   c0 = e00*(1-dx) + e10*dx ; c1 = e01*(1-dx) + e11*dx
    //   o  = c0*(1-dy) + c1*dy
    const v2f c0 = e00 * omx + e10 * dx;        // v_pk_mul + v_pk_fma
    const v2f c1 = e01 * omx + e11 * dx;        // v_pk_mul + v_pk_fma
    o[i]         = c0 * omy + c1 * dy;          // v_pk_mul + v_pk_fma
  }

  // 32 floats = 128 contiguous bytes per point: 8 x b128 non-temporal stores.
  // NT keeps the 256 MB output stream from evicting the tables out of L2.
  v4f*       op = out + (size_t)p * (kOutF / 4);
  const v4f* ov = (const v4f*)o;
#pragma unroll
  for (int k = 0; k < kOutF / 4; ++k) {
    __builtin_nontemporal_store(ov[k], op + k);
  }
}

extern "C" void kernel_launch(void* const* d_in, const int* in_sizes, int n_in,
                              void* d_out, int out_size, void* d_ws, size_t ws_size,
                              hipStream_t stream) {
  const float2* x      = (const float2*)d_in[0];   // (N_POINTS, 2) f32
  const v2f*    tables = (const v2f*)   d_in[1];   // (16, 2^19, 2) f32
  const int n = in_sizes[0] / 2;                   // N_POINTS

  const int threads = 256;                         // 8 wave32 per block
  const int blocks  = (n + threads - 1) / threads;
  hashgrid_encode_kernel<<<blocks, threads, 0, stream>>>(
      x, tables, (v4f*)d_out, n);
}